// KNNCrossAttentionBlock_22668837388565
// MI455X (gfx1250) — compile-verified
//
#include <hip/hip_runtime.h>
#include <hip/hip_bf16.h>
#include <math.h>

#define CCH 256           // channels (compile-time K and N of all GEMMs)
#define KNB 16            // neighbors
#define BB  2             // batch
#define NN  8192          // points per batch
#define BN  (BB*NN)       // 16384

typedef __attribute__((ext_vector_type(2))) float v2f;
typedef __attribute__((ext_vector_type(8))) float v8f;

// ---------------------------------------------------------------------------
// wave32 reduction
__device__ inline float wave_red_sum(float v) {
    #pragma unroll
    for (int off = 16; off >= 1; off >>= 1) v += __shfl_xor(v, off, 32);
    return v;
}

// top-16 register heap: replace current max if d is smaller (order irrelevant:
// all downstream uses are permutation-invariant reductions over k)
__device__ inline void topk_insert(float (&best)[16], int (&bid)[16], float &wv,
                                   float d, int id) {
    if (d < wv) {
        int slot = 0; float mv = best[0];
        #pragma unroll
        for (int t = 1; t < 16; t++) if (best[t] > mv) { mv = best[t]; slot = t; }
        #pragma unroll
        for (int t = 0; t < 16; t++) if (t == slot) { best[t] = d; bid[t] = id; }
        float w = best[0];
        #pragma unroll
        for (int t = 1; t < 16; t++) w = fmaxf(w, best[t]);
        wv = w;
    }
}

// ---------------------------------------------------------------------------
// f32 WMMA GEMM, K = N = 256 compile-time:  C[M,256] = A[M,256] * B (+bias).
// BT=0: B is [256,256] row-major; BT=1: B is [256,256] used as B^T.
// One wave per 16x16 tile; k-loop in chunks of 8 (8 b64 loads then 8 WMMAs)
// so loads overlap the WMMA pipe instead of s_wait_loadcnt 0 per step.
template <int BT>
__global__ void __launch_bounds__(32, 1)
gemm_wmma_f32(const float* __restrict__ A,
              const float* __restrict__ Bm,
              const float* __restrict__ bias,
              float* __restrict__ Cc) {
    const int lane = threadIdx.x & 31;
    const int r = lane & 15, half = lane >> 4;
    const int m0 = blockIdx.y * 16;
    const int n0 = blockIdx.x * 16;
    const float* arow = A + (size_t)(m0 + r) * CCH + 2 * half;
    const float* brow = BT ? (Bm + (size_t)(n0 + r) * CCH + 2 * half)
                           : (Bm + (size_t)(2 * half) * CCH + n0 + r);
    v8f acc = {};
    #pragma unroll
    for (int blk = 0; blk < 8; blk++) {
        v2f a[8], b[8];
        #pragma unroll
        for (int t = 0; t < 8; t++) {
            const int s = blk * 8 + t;
            a[t].x = arow[s * 4];
            a[t].y = arow[s * 4 + 1];
            if (BT) {
                b[t].x = brow[s * 4];
                b[t].y = brow[s * 4 + 1];
            } else {
                b[t].x = brow[(size_t)(s * 4) * CCH];
                b[t].y = brow[(size_t)(s * 4 + 1) * CCH];
            }
        }
        #pragma unroll
        for (int t = 0; t < 8; t++)
            acc = __builtin_amdgcn_wmma_f32_16x16x4_f32(false, a[t], false, b[t],
                                                        (short)0, acc, false, false);
    }
    #pragma unroll
    for (int i = 0; i < 8; i++) {
        const int row = m0 + i + 8 * half;
        const int col = n0 + r;
        float v = acc[i];
        if (bias) v += bias[col];
        Cc[(size_t)row * CCH + col] = v;
    }
}

// ---------------------------------------------------------------------------
// 3D brute-force KNN: one thread per query, candidates tiled through LDS.
__global__ void knn3d_kernel(const float* __restrict__ qpos,
                             const float* __restrict__ cpos,
                             int* __restrict__ idx) {
    const int CH = 1024;
    __shared__ float sp[CH * 3];
    const int q = blockIdx.x * blockDim.x + threadIdx.x;   // global query
    const int b = q >> 13;                                 // q / 8192
    const float qx = qpos[q * 3 + 0], qy = qpos[q * 3 + 1], qz = qpos[q * 3 + 2];
    float best[16]; int bid[16];
    #pragma unroll
    for (int t = 0; t < 16; t++) { best[t] = 3.4e38f; bid[t] = 0; }
    float wv = 3.4e38f;
    const float* cb = cpos + (size_t)b * NN * 3;
    for (int c0 = 0; c0 < NN; c0 += CH) {
        __syncthreads();
        for (int t = threadIdx.x; t < CH * 3; t += blockDim.x)
            sp[t] = cb[(size_t)c0 * 3 + t];
        __syncthreads();
        for (int j = 0; j < CH; j++) {
            const float dx = qx - sp[j * 3 + 0];
            const float dy = qy - sp[j * 3 + 1];
            const float dz = qz - sp[j * 3 + 2];
            topk_insert(best, bid, wv, dx * dx + dy * dy + dz * dz, c0 + j);
        }
    }
    #pragma unroll
    for (int t = 0; t < 16; t++) idx[(size_t)q * 16 + t] = bid[t];
}

// ---------------------------------------------------------------------------
// Fused feature-space KNN: per wave, a 16-query tile; stream all 8192
// candidates in 16-col tiles; 16x16x256 dot via 64 chained WMMA f32 ops
// (B loads chunked 8-wide for latency overlap); d2 = |a|^2+|b|^2-2ab;
// online top-16 per row via LDS roundtrip.  One wave per SIMD: ~200 live
// VGPRs (A-fragments resident) — launch_bounds prevents scratch spills.
__global__ void __launch_bounds__(128, 1)
knn_feat_kernel(const float* __restrict__ F,
                const float* __restrict__ sq,
                int* __restrict__ idx) {
    __shared__ float tile[4][16][17];
    const int lane = threadIdx.x & 31;
    const int w = threadIdx.x >> 5;
    const int mt = blockIdx.x * 4 + w;        // 16-row tile id (global)
    const int m0 = mt * 16;
    const int b = m0 >> 13;
    const int r = lane & 15, half = lane >> 4;
    const float* Fb = F + (size_t)b * NN * CCH;
    const int ml = m0 - b * NN;               // local row base

    // hoist the A fragments (this wave's 16 query rows) into registers
    const float* arow = Fb + (size_t)(ml + r) * CCH + 2 * half;
    v2f afrag[64];
    #pragma unroll
    for (int s = 0; s < 64; s++) {
        afrag[s].x = arow[s * 4];
        afrag[s].y = arow[s * 4 + 1];
    }
    // per-row |a|^2 (invariant over candidate loop)
    float sqa[8];
    #pragma unroll
    for (int i = 0; i < 8; i++) sqa[i] = sq[m0 + i + 8 * half];

    float best[16]; int bid[16];
    #pragma unroll
    for (int t = 0; t < 16; t++) { best[t] = 3.4e38f; bid[t] = 0; }
    float wv = 3.4e38f;

    for (int nn0 = 0; nn0 < NN; nn0 += 16) {
        const float* brow = Fb + (size_t)(nn0 + r) * CCH + 2 * half;
        v8f acc = {};
        #pragma unroll
        for (int blk = 0; blk < 8; blk++) {
            v2f bf[8];
            #pragma unroll
            for (int t = 0; t < 8; t++) {
                const int s = blk * 8 + t;     // B[k][n] = F[nn0+n][k]
                bf[t].x = brow[s * 4];
                bf[t].y = brow[s * 4 + 1];
            }
            #pragma unroll
            for (int t = 0; t < 8; t++)
                acc = __builtin_amdgcn_wmma_f32_16x16x4_f32(false, afrag[blk * 8 + t],
                                                            false, bf[t],
                                                            (short)0, acc, false, false);
        }
        const float sqb = sq[b * NN + nn0 + r];
        __syncthreads();
        #pragma unroll
        for (int i = 0; i < 8; i++)
            tile[w][i + 8 * half][r] = sqa[i] + sqb - 2.0f * acc[i];
        __syncthreads();
        if (lane < 16) {
            #pragma unroll
            for (int j = 0; j < 16; j++)
                topk_insert(best, bid, wv, tile[w][lane][j], nn0 + j);
        }
    }
    if (lane < 16) {
        #pragma unroll
        for (int t = 0; t < 16; t++) idx[(size_t)(m0 + lane) * 16 + t] = bid[t];
    }
}

// ---------------------------------------------------------------------------
// row squared norms: one wave per row
__global__ void rownorm_kernel(const float* __restrict__ F, float* __restrict__ sq) {
    const int lane = threadIdx.x & 31;
    const int n = blockIdx.x * 4 + (threadIdx.x >> 5);
    float s = 0.f;
    #pragma unroll
    for (int t = 0; t < 8; t++) {
        const float v = F[(size_t)n * CCH + lane * 8 + t];
        s += v * v;
    }
    s = wave_red_sum(s);
    if (lane == 0) sq[n] = s;
}

// ---------------------------------------------------------------------------
// prep round 1: U1 = curr_pos@Wp0 + bp0 ; G1 = prev - prev_pos@Wp0
__global__ void prep1_kernel(const float* __restrict__ cpos,
                             const float* __restrict__ ppos,
                             const float* __restrict__ prev,
                             const float* __restrict__ Wp0,
                             const float* __restrict__ bp0,
                             float* __restrict__ U1, float* __restrict__ G1) {
    const size_t i = (size_t)blockIdx.x * blockDim.x + threadIdx.x;  // BN*C
    const int n = (int)(i >> 8), c = (int)(i & 255);
    const float w0 = Wp0[c], w1 = Wp0[CCH + c], w2 = Wp0[2 * CCH + c];
    U1[i] = bp0[c] + cpos[n * 3] * w0 + cpos[n * 3 + 1] * w1 + cpos[n * 3 + 2] * w2;
    G1[i] = prev[i] - (ppos[n * 3] * w0 + ppos[n * 3 + 1] * w1 + ppos[n * 3 + 2] * w2);
}

// prep round 2: U2 = P + bp1 (in place over P) ; G2 = F0 - P
__global__ void prep2_kernel(const float* __restrict__ P,
                             const float* __restrict__ F0,
                             const float* __restrict__ bp1,
                             float* __restrict__ U2, float* __restrict__ G2) {
    const size_t i = (size_t)blockIdx.x * blockDim.x + threadIdx.x;
    const int c = (int)(i & 255);
    const float p = P[i];
    U2[i] = p + bp1[c];
    G2[i] = F0[i] - p;
}

// ---------------------------------------------------------------------------
// Fused attention: tgt[n,k] = U[n]+G[j_k]; logit = (U+G_j)·QK[n]/16;
// S[n] = U[n] + sum_k softmax_k * G[j_k].  One wave per point, online softmax.
__global__ void attn_fused_kernel(const float* __restrict__ U,
                                  const float* __restrict__ G,
                                  const float* __restrict__ QK,
                                  const int* __restrict__ idx,
                                  float* __restrict__ S) {
    const int lane = threadIdx.x & 31;
    const int n = blockIdx.x * 4 + (threadIdx.x >> 5);
    const int b = n >> 13;
    const int c0 = lane * 8;
    float u[8], q[8];
    float bp = 0.f;
    #pragma unroll
    for (int t = 0; t < 8; t++) {
        u[t] = U[(size_t)n * CCH + c0 + t];
        q[t] = QK[(size_t)n * CCH + c0 + t];
        bp += u[t] * q[t];
    }
    const float base = wave_red_sum(bp);
    float m = -3.4e38f, denom = 0.f;
    float acc[8];
    #pragma unroll
    for (int t = 0; t < 8; t++) acc[t] = 0.f;
    for (int k = 0; k < KNB; k++) {
        const int j = idx[(size_t)n * KNB + k] + b * NN;
        float g[8], dp = 0.f;
        #pragma unroll
        for (int t = 0; t < 8; t++) {
            g[t] = G[(size_t)j * CCH + c0 + t];
            dp += g[t] * q[t];
        }
        dp = wave_red_sum(dp);
        const float logit = (base + dp) * (1.0f / 16.0f);   // 1/sqrt(256)
        const float nm = fmaxf(m, logit);
        const float scale = __expf(m - nm);
        const float e = __expf(logit - nm);
        denom = denom * scale + e;
        #pragma unroll
        for (int t = 0; t < 8; t++) acc[t] = acc[t] * scale + e * g[t];
        m = nm;
    }
    const float inv = 1.0f / denom;
    #pragma unroll
    for (int t = 0; t < 8; t++)
        S[(size_t)n * CCH + c0 + t] = u[t] + acc[t] * inv;
}

// ---------------------------------------------------------------------------
// T0 = LayerNorm(Xa + Xb) * g + b        (one wave per row)
__global__ void ln_add_kernel(const float* __restrict__ Xa,
                              const float* __restrict__ Xb,
                              const float* __restrict__ g,
                              const float* __restrict__ bt,
                              float* __restrict__ O) {
    const int lane = threadIdx.x & 31;
    const int n = blockIdx.x * 4 + (threadIdx.x >> 5);
    const int c0 = lane * 8;
    float x[8], s = 0.f;
    #pragma unroll
    for (int t = 0; t < 8; t++) {
        x[t] = Xa[(size_t)n * CCH + c0 + t] + Xb[(size_t)n * CCH + c0 + t];
        s += x[t];
    }
    const float mu = wave_red_sum(s) * (1.0f / CCH);
    float v = 0.f;
    #pragma unroll
    for (int t = 0; t < 8; t++) { const float d = x[t] - mu; v += d * d; }
    const float var = wave_red_sum(v) * (1.0f / CCH);
    const float is = rsqrtf(var + 1e-5f);
    #pragma unroll
    for (int t = 0; t < 8; t++)
        O[(size_t)n * CCH + c0 + t] = (x[t] - mu) * is * g[c0 + t] + bt[c0 + t];
}

// out[(nl*B + b)*C + c] = LayerNorm(curr + T1)    ([N,B,C] transpose)
__global__ void ln_final_kernel(const float* __restrict__ curr,
                                const float* __restrict__ T1,
                                const float* __restrict__ g,
                                const float* __restrict__ bt,
                                float* __restrict__ out) {
    const int lane = threadIdx.x & 31;
    const int n = blockIdx.x * 4 + (threadIdx.x >> 5);
    const int c0 = lane * 8;
    float x[8], s = 0.f;
    #pragma unroll
    for (int t = 0; t < 8; t++) {
        x[t] = curr[(size_t)n * CCH + c0 + t] + T1[(size_t)n * CCH + c0 + t];
        s += x[t];
    }
    const float mu = wave_red_sum(s) * (1.0f / CCH);
    float v = 0.f;
    #pragma unroll
    for (int t = 0; t < 8; t++) { const float d = x[t] - mu; v += d * d; }
    const float var = wave_red_sum(v) * (1.0f / CCH);
    const float is = rsqrtf(var + 1e-5f);
    const int b = n >> 13, nl = n & (NN - 1);
    #pragma unroll
    for (int t = 0; t < 8; t++)
        out[((size_t)nl * BB + b) * CCH + c0 + t] =
            (x[t] - mu) * is * g[c0 + t] + bt[c0 + t];
}

// ---------------------------------------------------------------------------
extern "C" void kernel_launch(void* const* d_in, const int* in_sizes, int n_in,
                              void* d_out, int out_size, void* d_ws, size_t ws_size,
                              hipStream_t stream) {
    (void)in_sizes; (void)n_in; (void)out_size; (void)ws_size;
    const float* prev  = (const float*)d_in[0];
    const float* curr  = (const float*)d_in[1];
    const float* ppos  = (const float*)d_in[2];
    const float* cpos  = (const float*)d_in[3];
    const float* Wq0   = (const float*)d_in[4];
    const float* Wk0   = (const float*)d_in[5];
    const float* Wv0   = (const float*)d_in[6];
    const float* Wq1   = (const float*)d_in[7];
    const float* Wk1   = (const float*)d_in[8];
    const float* Wv1   = (const float*)d_in[9];
    const float* Wp0   = (const float*)d_in[10];
    const float* bp0   = (const float*)d_in[11];
    const float* Wp1   = (const float*)d_in[12];
    const float* bp1   = (const float*)d_in[13];
    const float* Wl    = (const float*)d_in[14];
    const float* bl    = (const float*)d_in[15];
    const float* g0    = (const float*)d_in[16];
    const float* b0    = (const float*)d_in[17];
    const float* g1    = (const float*)d_in[18];
    const float* b1    = (const float*)d_in[19];

    float* ws = (float*)d_ws;
    size_t o = 0;
    float* Wqk0 = ws + o; o += CCH * CCH;
    float* Wqk1 = ws + o; o += CCH * CCH;
    float* sq   = ws + o; o += BN;
    int*   idx1 = (int*)(ws + o); o += (size_t)BN * KNB;
    int*   idx2 = (int*)(ws + o); o += (size_t)BN * KNB;
    const size_t BIG = (size_t)BN * CCH;
    float* bufA = ws + o; o += BIG;   // QK0 -> QK1 -> A2
    float* bufB = ws + o; o += BIG;   // U1  -> P/U2 -> T1
    float* bufC = ws + o; o += BIG;   // G1  -> G2
    float* bufD = ws + o; o += BIG;   // S1  -> S2 -> T0
    float* bufE = ws + o; o += BIG;   // F0

    const dim3 gW(CCH / 16, CCH / 16);     // 256x256 GEMM tiles
    const dim3 gB(CCH / 16, BN / 16);      // 16384x256 GEMM tiles
    const dim3 blk1w(32);

    // weight products Wq*Wk^T
    gemm_wmma_f32<1><<<gW, blk1w, 0, stream>>>(Wq0, Wk0, nullptr, Wqk0);
    gemm_wmma_f32<1><<<gW, blk1w, 0, stream>>>(Wq1, Wk1, nullptr, Wqk1);
    // QK0 = curr @ Wqk0
    gemm_wmma_f32<0><<<gB, blk1w, 0, stream>>>(curr, Wqk0, nullptr, bufA);
    // U1/G1
    prep1_kernel<<<BN, 256, 0, stream>>>(cpos, ppos, prev, Wp0, bp0, bufB, bufC);
    // 3D KNN
    knn3d_kernel<<<BN / 256, 256, 0, stream>>>(cpos, ppos, idx1);
    // attention round 1 -> S1
    attn_fused_kernel<<<BN / 4, 128, 0, stream>>>(bufB, bufC, bufA, idx1, bufD);
    // F0 = S1 @ Wv0
    gemm_wmma_f32<0><<<gB, blk1w, 0, stream>>>(bufD, Wv0, nullptr, bufE);
    // feature-space KNN
    rownorm_kernel<<<BN / 4, 128, 0, stream>>>(bufE, sq);
    knn_feat_kernel<<<BN / 64, 128, 0, stream>>>(bufE, sq, idx2);
    // QK1 = F0 @ Wqk1 ; P = F0 @ Wp1
    gemm_wmma_f32<0><<<gB, blk1w, 0, stream>>>(bufE, Wqk1, nullptr, bufA);
    gemm_wmma_f32<0><<<gB, blk1w, 0, stream>>>(bufE, Wp1, nullptr, bufB);
    prep2_kernel<<<BN, 256, 0, stream>>>(bufB, bufE, bp1, bufB, bufC);
    // attention round 2 -> S2
    attn_fused_kernel<<<BN / 4, 128, 0, stream>>>(bufB, bufC, bufA, idx2, bufD);
    // A2 = S2 @ Wv1
    gemm_wmma_f32<0><<<gB, blk1w, 0, stream>>>(bufD, Wv1, nullptr, bufA);
    // T0 = LN(curr + A2)
    ln_add_kernel<<<BN / 4, 128, 0, stream>>>(curr, bufA, g0, b0, bufD);
    // T1 = T0 @ Wl + bl
    gemm_wmma_f32<0><<<gB, blk1w, 0, stream>>>(bufD, Wl, bl, bufB);
    // out = LN(curr + T1), [N,B,C]
    ln_final_kernel<<<BN / 4, 128, 0, stream>>>(curr, bufB, g1, b1, (float*)d_out);
}